// MultiHeadGraphAttention_77232101916982
// MI455X (gfx1250) — compile-verified
//
#include <hip/hip_runtime.h>
#include <hip/hip_bf16.h>
#include <math.h>

#define N_NODES 100000
#define N_EDGES 1600000
#define D_IN    128
#define HEADS   8
#define UNITS   16
#define D_OUT   128            // HEADS*UNITS

typedef float v2f __attribute__((ext_vector_type(2)));
typedef float v8f __attribute__((ext_vector_type(8)));

#define BT_STRIDE 132          // transposed-B LDS row stride (floats); bank-conflict-free b64 reads
#define M_TILES (N_NODES / 16) // 6250 exactly
#define WAVES_PER_BLOCK 8

// ---------------- monotone float <-> u32 order encoding (for atomic max) ---------
__device__ __forceinline__ unsigned f2ord(float f) {
  unsigned u = __float_as_uint(f);
  return (u & 0x80000000u) ? ~u : (u | 0x80000000u);
}
__device__ __forceinline__ float ord2f(unsigned u) {
  u = (u & 0x80000000u) ? (u & 0x7fffffffu) : ~u;
  return __uint_as_float(u);
}

// ---------------- 1) xp = x @ kernel via V_WMMA_F32_16X16X4_F32 ------------------
// One wave computes a 16x128 strip of xp. B (128x128) staged TRANSPOSED in LDS:
//   sBt[n * BT_STRIDE + k] = kernel[k][n]
// so each lane's B fragment {K=k0+koff, k0+koff+1} for column n is one ds_load_b64.
// Fragment layouts per CDNA5 ISA 7.12.2:
//   A 16x4 : lanes 0-15 -> M=lane, {K=k0,k0+1}; lanes 16-31 -> M=lane-16, {K=k0+2,k0+3}
//   B 4x16 : lanes 0-15 -> N=lane, {K=k0,k0+1}; lanes 16-31 -> N=lane-16, {K=k0+2,k0+3}
//   C/D    : VGPR r, lanes 0-15 -> M=r, N=lane; lanes 16-31 -> M=8+r, N=lane-16
__global__ void gat_xp_wmma(const float* __restrict__ x,
                            const float* __restrict__ kmat,
                            float* __restrict__ xp) {
  extern __shared__ float sBt[];
  for (int idx = threadIdx.x; idx < D_IN * D_OUT; idx += blockDim.x) {
    const int k = idx >> 7, n = idx & 127;
    sBt[n * BT_STRIDE + k] = kmat[idx];
  }
  __syncthreads();

  const int wave = threadIdx.x >> 5;
  const int lane = threadIdx.x & 31;
  const int tile = blockIdx.x * WAVES_PER_BLOCK + wave;
  if (tile >= M_TILES) return;

  const int rowBase = tile * 16;
  const int m    = lane & 15;
  const int half = lane >> 4;   // 0 -> K+{0,1}, 1 -> K+{2,3}
  const int koff = half * 2;

  const float2* xr2 = reinterpret_cast<const float2*>(x + (size_t)(rowBase + m) * D_IN);

  v8f acc[8] = {};
  v2f bcur[8], bnxt[8];

  // prologue: fragments for kk = 0
  float2 avf = xr2[half];
#pragma unroll
  for (int t = 0; t < 8; ++t) {
    const float2 bv = *reinterpret_cast<const float2*>(
        &sBt[(t * 16 + m) * BT_STRIDE + koff]);
    bcur[t].x = bv.x; bcur[t].y = bv.y;
  }

#pragma unroll
  for (int kk = 0; kk < D_IN / 4; ++kk) {
    v2f a; a.x = avf.x; a.y = avf.y;

    // software pipeline: issue next-step loads before this step's WMMAs
    if (kk < D_IN / 4 - 1) {
      avf = xr2[(kk + 1) * 2 + half];
#pragma unroll
      for (int t = 0; t < 8; ++t) {
        const float2 bv = *reinterpret_cast<const float2*>(
            &sBt[(t * 16 + m) * BT_STRIDE + (kk + 1) * 4 + koff]);
        bnxt[t].x = bv.x; bnxt[t].y = bv.y;
      }
    }

#pragma unroll
    for (int t = 0; t < 8; ++t) {
      acc[t] = __builtin_amdgcn_wmma_f32_16x16x4_f32(
          false, a, false, bcur[t], (short)0, acc[t], false, false);
    }

#pragma unroll
    for (int t = 0; t < 8; ++t) bcur[t] = bnxt[t];
  }

  const int rowHalf = half * 8;
#pragma unroll
  for (int t = 0; t < 8; ++t) {
#pragma unroll
    for (int r = 0; r < 8; ++r) {
      xp[(size_t)(rowBase + rowHalf + r) * D_OUT + t * 16 + m] = acc[t][r];
    }
  }
}

// ---------------- 2) f_t[n,h] = <xp[n,h,:], ka1[h,:]>, f_s likewise --------------
__global__ void gat_attn_feat(const float* __restrict__ xp,
                              const float* __restrict__ ka1,
                              const float* __restrict__ ka2,
                              float* __restrict__ f_t,
                              float* __restrict__ f_s) {
  int gid = blockIdx.x * blockDim.x + threadIdx.x;
  if (gid >= N_NODES * HEADS) return;
  const int n = gid >> 3, h = gid & 7;
  const float* p  = xp + (size_t)n * D_OUT + h * UNITS;
  const float* a1 = ka1 + h * UNITS;
  const float* a2 = ka2 + h * UNITS;
  float ft = 0.f, fs = 0.f;
#pragma unroll
  for (int u = 0; u < UNITS; ++u) { float v = p[u]; ft += v * a1[u]; fs += v * a2[u]; }
  f_t[gid] = ft;
  f_s[gid] = fs;
}

// ---------------- 3) zero-init accumulators --------------------------------------
__global__ void gat_init(unsigned* __restrict__ smax,
                         float* __restrict__ denom,
                         float* __restrict__ out) {
  int gid = blockIdx.x * blockDim.x + threadIdx.x;
  if (gid < N_NODES * HEADS) { smax[gid] = 0u; denom[gid] = 0.f; }
  if (gid < N_NODES * D_OUT) out[gid] = 0.f;
}

// ---------------- 4) segment max of leaky_relu scores ----------------------------
__global__ void gat_edge_max(const long long* __restrict__ edges,
                             const float* __restrict__ f_t,
                             const float* __restrict__ f_s,
                             unsigned* __restrict__ smax) {
  int gid = blockIdx.x * blockDim.x + threadIdx.x;
  if (gid >= N_EDGES * HEADS) return;
  const int e = gid >> 3, h = gid & 7;
  const int src = (int)edges[2 * (long long)e];
  const int tgt = (int)edges[2 * (long long)e + 1];
  float s = f_t[tgt * HEADS + h] + f_s[src * HEADS + h];
  s = (s > 0.f) ? s : 0.2f * s;
  atomicMax(smax + tgt * HEADS + h, f2ord(s));
}

// ---------------- 5) segment sum of exp(score - max) -----------------------------
__global__ void gat_edge_denom(const long long* __restrict__ edges,
                               const float* __restrict__ f_t,
                               const float* __restrict__ f_s,
                               const unsigned* __restrict__ smax,
                               float* __restrict__ denom) {
  int gid = blockIdx.x * blockDim.x + threadIdx.x;
  if (gid >= N_EDGES * HEADS) return;
  const int e = gid >> 3, h = gid & 7;
  const int src = (int)edges[2 * (long long)e];
  const int tgt = (int)edges[2 * (long long)e + 1];
  const int ti = tgt * HEADS + h;
  float s = f_t[ti] + f_s[src * HEADS + h];
  s = (s > 0.f) ? s : 0.2f * s;
  unsafeAtomicAdd(denom + ti, __expf(s - ord2f(smax[ti])));
}

// ---------------- 6) out[tgt] += w * xp[src]  (L2-resident gather + atomics) -----
__global__ void gat_edge_aggr(const long long* __restrict__ edges,
                              const float* __restrict__ f_t,
                              const float* __restrict__ f_s,
                              const unsigned* __restrict__ smax,
                              const float* __restrict__ denom,
                              const float* __restrict__ xp,
                              float* __restrict__ out) {
  int gid = blockIdx.x * blockDim.x + threadIdx.x;
  if (gid >= N_EDGES * HEADS) return;
  const int e = gid >> 3, h = gid & 7;
  const int src = (int)edges[2 * (long long)e];
  const int tgt = (int)edges[2 * (long long)e + 1];

  const float* xr = xp + (size_t)src * D_OUT + h * UNITS;
  __builtin_prefetch(xr, 0, 3);   // global_prefetch_b8: start the gather early

  const int ti = tgt * HEADS + h;
  float s = f_t[ti] + f_s[src * HEADS + h];
  s = (s > 0.f) ? s : 0.2f * s;
  const float w = __expf(s - ord2f(smax[ti])) / (denom[ti] + 1e-7f);

  const float4* xv = reinterpret_cast<const float4*>(xr);
  const float4 p0 = xv[0], p1 = xv[1], p2 = xv[2], p3 = xv[3];
  float* op = out + (size_t)tgt * D_OUT + h * UNITS;
  unsafeAtomicAdd(op + 0,  w * p0.x);
  unsafeAtomicAdd(op + 1,  w * p0.y);
  unsafeAtomicAdd(op + 2,  w * p0.z);
  unsafeAtomicAdd(op + 3,  w * p0.w);
  unsafeAtomicAdd(op + 4,  w * p1.x);
  unsafeAtomicAdd(op + 5,  w * p1.y);
  unsafeAtomicAdd(op + 6,  w * p1.z);
  unsafeAtomicAdd(op + 7,  w * p1.w);
  unsafeAtomicAdd(op + 8,  w * p2.x);
  unsafeAtomicAdd(op + 9,  w * p2.y);
  unsafeAtomicAdd(op + 10, w * p2.z);
  unsafeAtomicAdd(op + 11, w * p2.w);
  unsafeAtomicAdd(op + 12, w * p3.x);
  unsafeAtomicAdd(op + 13, w * p3.y);
  unsafeAtomicAdd(op + 14, w * p3.z);
  unsafeAtomicAdd(op + 15, w * p3.w);
}

// ---------------- 7) out = elu(out + bias) ---------------------------------------
__global__ void gat_final(float* __restrict__ out, const float* __restrict__ bias) {
  int gid = blockIdx.x * blockDim.x + threadIdx.x;
  if (gid >= N_NODES * D_OUT) return;
  float v = out[gid] + bias[gid & (D_OUT - 1)];
  out[gid] = (v > 0.f) ? v : expm1f(v);
}

extern "C" void kernel_launch(void* const* d_in, const int* in_sizes, int n_in,
                              void* d_out, int out_size, void* d_ws, size_t ws_size,
                              hipStream_t stream) {
  const float*     x     = (const float*)d_in[0];
  const long long* edges = (const long long*)d_in[1];   // int64 edges
  const float*     kmat  = (const float*)d_in[2];
  const float*     ka1   = (const float*)d_in[3];
  const float*     ka2   = (const float*)d_in[4];
  const float*     bias  = (const float*)d_in[5];
  float*           out   = (float*)d_out;

  // workspace layout (≈61 MB)
  float*    xp    = (float*)d_ws;
  float*    f_t   = xp + (size_t)N_NODES * D_OUT;
  float*    f_s   = f_t + (size_t)N_NODES * HEADS;
  unsigned* smax  = (unsigned*)(f_s + (size_t)N_NODES * HEADS);
  float*    denom = (float*)(smax + (size_t)N_NODES * HEADS);

  const int gemmBlocks = (M_TILES + WAVES_PER_BLOCK - 1) / WAVES_PER_BLOCK;
  const size_t ldsB = (size_t)D_OUT * BT_STRIDE * sizeof(float);  // 67584 B

  gat_xp_wmma<<<gemmBlocks, 32 * WAVES_PER_BLOCK, ldsB, stream>>>(x, kmat, xp);
  gat_attn_feat<<<(N_NODES * HEADS + 255) / 256, 256, 0, stream>>>(xp, ka1, ka2, f_t, f_s);
  gat_init<<<(N_NODES * D_OUT + 255) / 256, 256, 0, stream>>>(smax, denom, out);
  gat_edge_max<<<(N_EDGES * HEADS + 255) / 256, 256, 0, stream>>>(edges, f_t, f_s, smax);
  gat_edge_denom<<<(N_EDGES * HEADS + 255) / 256, 256, 0, stream>>>(edges, f_t, f_s, smax, denom);
  gat_edge_aggr<<<(N_EDGES * HEADS + 255) / 256, 256, 0, stream>>>(edges, f_t, f_s, smax, denom, xp, out);
  gat_final<<<(N_NODES * D_OUT + 255) / 256, 256, 0, stream>>>(out, bias);
}